// ManoLayer_78932908966337
// MI455X (gfx1250) — compile-verified
//
#include <hip/hip_runtime.h>

typedef float v2f __attribute__((ext_vector_type(2)));
typedef float v8f __attribute__((ext_vector_type(8)));

#define V_    778
#define VPAD  784            // 49 * 16
#define J_    16
#define NB_   10
#define NP_   135
#define NC_   45
#define K_    148            // 135 pose_map + 10 betas + 1 template + 2 pad
#define NCOL  (VPAD * 3)     // 2352 padded columns
#define KSTEPS (K_ / 4)      // 37 WMMA k-steps

__device__ __constant__ int c_TIPS[5]  = {745, 317, 444, 556, 673};
__device__ __constant__ int c_ORDER[21] = {0,13,14,15,16,1,2,3,17,4,5,6,18,10,11,12,19,7,8,9,20};

// ---------------------------------------------------------------------------
// Build combined direction matrix Dir[k][n]:
//   k in [0,135)  : posedirs[v][c][k]
//   k in [135,145): shapedirs[v][c][k-135]
//   k == 145      : v_template[v][c]
//   else 0;  n = v*3 + c, zero-padded for v >= 778.
// ---------------------------------------------------------------------------
__global__ void prep_dirs(const float* __restrict__ posedirs,
                          const float* __restrict__ shapedirs,
                          const float* __restrict__ vtemplate,
                          float* __restrict__ Dir) {
    int idx = blockIdx.x * blockDim.x + threadIdx.x;
    if (idx >= K_ * NCOL) return;
    int k = idx / NCOL, n = idx % NCOL;
    int v = n / 3, c = n % 3;
    float val = 0.f;
    if (v < V_) {
        if (k < 135)       val = posedirs[(v * 3 + c) * NP_ + k];
        else if (k < 145)  val = shapedirs[(v * 3 + c) * NB_ + (k - 135)];
        else if (k == 145) val = vtemplate[v * 3 + c];
    }
    Dir[k * NCOL + n] = val;
}

// ---------------------------------------------------------------------------
// Reduced joint regressor: JtJsd[j][c][s] ; s<10 -> JR@shapedirs, s==10 -> JR@template
// ---------------------------------------------------------------------------
__global__ void prep_jreg(const float* __restrict__ JR,
                          const float* __restrict__ shapedirs,
                          const float* __restrict__ vtemplate,
                          float* __restrict__ JtJsd) {
    int t = threadIdx.x;
    if (t >= J_ * 3 * 11) return;
    int joint = t / 33, rem = t % 33, c = rem / 11, s = rem % 11;
    float acc = 0.f;
    for (int v = 0; v < V_; ++v) {
        float w = JR[joint * V_ + v];
        acc += w * (s < 10 ? shapedirs[(v * 3 + c) * NB_ + s] : vtemplate[v * 3 + c]);
    }
    JtJsd[t] = acc;
}

// ---------------------------------------------------------------------------
// Rodrigues (matches reference: angle = sqrt(dot + 1e-8))
// ---------------------------------------------------------------------------
__device__ inline void rodrigues(float ax, float ay, float az, float* R) {
    float sq    = ax * ax + ay * ay + az * az + 1e-8f;
    float angle = sqrtf(sq);
    float inv   = 1.f / angle;
    float x = ax * inv, y = ay * inv, z = az * inv;
    float c = cosf(angle), s = sinf(angle), t = 1.f - c;
    R[0] = 1.f + t * (-(y * y + z * z));
    R[1] = -s * z + t * x * y;
    R[2] =  s * y + t * x * z;
    R[3] =  s * z + t * x * y;
    R[4] = 1.f + t * (-(x * x + z * z));
    R[5] = -s * x + t * y * z;
    R[6] = -s * y + t * x * z;
    R[7] =  s * x + t * y * z;
    R[8] = 1.f + t * (-(x * x + y * y));
}

// ---------------------------------------------------------------------------
// Per-batch pose processing: coef vector, skinning transforms (3x4), joints.
// ---------------------------------------------------------------------------
__global__ void pose_kernel(const float* __restrict__ pc,
                            const float* __restrict__ betas,
                            const float* __restrict__ comps,
                            const float* __restrict__ hmean,
                            const float* __restrict__ JtJsd,
                            float* __restrict__ coef,
                            float* __restrict__ R2out,
                            float* __restrict__ jtr16,
                            int B) {
    int b = blockIdx.x * blockDim.x + threadIdx.x;
    if (b >= B) return;
    const float* p = pc + (size_t)b * 48;

    // full pose = [global(3), hands_mean + pose_pca @ selected_comps]
    float fp[48];
    fp[0] = p[0]; fp[1] = p[1]; fp[2] = p[2];
    for (int c = 0; c < 45; ++c) {
        float h = hmean[c];
        for (int k = 0; k < 45; ++k) h += p[3 + k] * comps[k * 45 + c];
        fp[3 + c] = h;
    }

    // rest joints j = Jt + Jsd @ betas
    float jnt[16][3];
    for (int jj = 0; jj < 16; ++jj)
        for (int c = 0; c < 3; ++c) {
            const float* row = JtJsd + (jj * 3 + c) * 11;
            float acc = row[10];
            #pragma unroll
            for (int s = 0; s < 10; ++s) acc += row[s] * betas[(size_t)b * NB_ + s];
            jnt[jj][c] = acc;
        }

    float* cf = coef + (size_t)b * K_;
    #pragma unroll
    for (int s = 0; s < 10; ++s) cf[135 + s] = betas[(size_t)b * NB_ + s];
    cf[145] = 1.f; cf[146] = 0.f; cf[147] = 0.f;

    // root transform
    float R0[9];
    rodrigues(fp[0], fp[1], fp[2], R0);
    float G0[12];
    #pragma unroll
    for (int r = 0; r < 3; ++r) {
        G0[r * 4 + 0] = R0[r * 3 + 0];
        G0[r * 4 + 1] = R0[r * 3 + 1];
        G0[r * 4 + 2] = R0[r * 3 + 2];
        G0[r * 4 + 3] = jnt[0][r];
    }
    {
        float* o = R2out + ((size_t)b * 16 + 0) * 12;
        #pragma unroll
        for (int r = 0; r < 3; ++r) {
            o[r * 4 + 0] = G0[r * 4 + 0];
            o[r * 4 + 1] = G0[r * 4 + 1];
            o[r * 4 + 2] = G0[r * 4 + 2];
            o[r * 4 + 3] = G0[r * 4 + 3] - (G0[r * 4 + 0] * jnt[0][0] +
                                            G0[r * 4 + 1] * jnt[0][1] +
                                            G0[r * 4 + 2] * jnt[0][2]);
        }
        float* jo = jtr16 + ((size_t)b * 16 + 0) * 3;
        jo[0] = G0[3]; jo[1] = G0[7]; jo[2] = G0[11];
    }

    // five depth-3 chains rooted at joint 0
    const int chainStart[5] = {1, 4, 7, 10, 13};
    for (int cc = 0; cc < 5; ++cc) {
        float Gp[12];
        #pragma unroll
        for (int i = 0; i < 12; ++i) Gp[i] = G0[i];
        float jp[3] = {jnt[0][0], jnt[0][1], jnt[0][2]};
        int k = chainStart[cc];
        for (int step = 0; step < 3; ++step, ++k) {
            float Rk[9];
            rodrigues(fp[3 * k], fp[3 * k + 1], fp[3 * k + 2], Rk);
            // pose_map entry (rots[1:] - I), row-major
            float* pm = cf + (k - 1) * 9;
            #pragma unroll
            for (int i = 0; i < 9; ++i) pm[i] = Rk[i] - ((i == 0 || i == 4 || i == 8) ? 1.f : 0.f);
            float tl[3] = {jnt[k][0] - jp[0], jnt[k][1] - jp[1], jnt[k][2] - jp[2]};
            float Gn[12];
            #pragma unroll
            for (int r = 0; r < 3; ++r) {
                float a0 = Gp[r * 4 + 0], a1 = Gp[r * 4 + 1], a2 = Gp[r * 4 + 2];
                Gn[r * 4 + 0] = a0 * Rk[0] + a1 * Rk[3] + a2 * Rk[6];
                Gn[r * 4 + 1] = a0 * Rk[1] + a1 * Rk[4] + a2 * Rk[7];
                Gn[r * 4 + 2] = a0 * Rk[2] + a1 * Rk[5] + a2 * Rk[8];
                Gn[r * 4 + 3] = a0 * tl[0] + a1 * tl[1] + a2 * tl[2] + Gp[r * 4 + 3];
            }
            float* o = R2out + ((size_t)b * 16 + k) * 12;
            #pragma unroll
            for (int r = 0; r < 3; ++r) {
                o[r * 4 + 0] = Gn[r * 4 + 0];
                o[r * 4 + 1] = Gn[r * 4 + 1];
                o[r * 4 + 2] = Gn[r * 4 + 2];
                o[r * 4 + 3] = Gn[r * 4 + 3] - (Gn[r * 4 + 0] * jnt[k][0] +
                                                Gn[r * 4 + 1] * jnt[k][1] +
                                                Gn[r * 4 + 2] * jnt[k][2]);
            }
            float* jo = jtr16 + ((size_t)b * 16 + k) * 3;
            jo[0] = Gn[3]; jo[1] = Gn[7]; jo[2] = Gn[11];
            #pragma unroll
            for (int i = 0; i < 12; ++i) Gp[i] = Gn[i];
            jp[0] = jnt[k][0]; jp[1] = jnt[k][1]; jp[2] = jnt[k][2];
        }
    }
}

// ---------------------------------------------------------------------------
// Main fused kernel, two WMMA stages per workgroup (3 waves, 96 threads):
//   Stage 1: v_posed tile   = coef(16b x 148) @ Dir(148 x 48)   [37 WMMA/wave]
//   Stage 2: blended LBS tf = W(16v x 16j) @ R2[b](16j x 12e)   [4 WMMA per b]
//   Stage 3: tiny VALU apply: vert = T(3x4) @ [v_posed, 1] * 1000
// ---------------------------------------------------------------------------
__global__ __launch_bounds__(96)
void mano_main(const float* __restrict__ Dir,
               const float* __restrict__ coef,
               const float* __restrict__ R2,
               const float* __restrict__ W,
               float* __restrict__ out_verts,
               int B) {
    __shared__ float s_coef[16][K_];    // batch-tile coefficient vectors
    __shared__ float s_R2[16][192];     // 16 batches x 16 joints x 12
    __shared__ float s_w[16][16];       // 16 vertices x 16 joints
    __shared__ float s_vp[16][48];      // v_posed tile: [batch][local n]
    __shared__ float s_T[16][16][17];   // blended transforms [batch][vert][e], padded

    int vt = blockIdx.x;               // vertex tile (0..48)
    int bt = blockIdx.y;               // batch tile
    int tid = threadIdx.x;
    int wave = tid >> 5, lane = tid & 31;

    for (int i = tid; i < 16 * K_; i += 96) {
        int bm = i / K_, kk = i % K_;
        int b = bt * 16 + bm;
        s_coef[bm][kk] = (b < B) ? coef[(size_t)b * K_ + kk] : 0.f;
    }
    for (int i = tid; i < 16 * 192; i += 96) {
        int bm = i / 192, e = i % 192;
        int b = bt * 16 + bm;
        s_R2[bm][e] = (b < B) ? R2[(size_t)b * 192 + e] : 0.f;
    }
    for (int i = tid; i < 256; i += 96) {
        int vm = i / 16, jj = i % 16;
        int v = vt * 16 + vm;
        s_w[vm][jj] = (v < V_) ? W[v * J_ + jj] : 0.f;
    }
    __syncthreads();

    // fragment lane mappings (ISA 7.12.2, f32 16x16x4)
    int fm = lane & 15;                // A: M = lane%16 / B: N = lane%16
    int fk = (lane < 16) ? 0 : 2;      // first K held by this lane half

    // --- Stage 1: WMMA f32 GEMM, K = 148 (37 steps) ---
    int n0 = vt * 48 + wave * 16;
    const float* dcol = Dir + n0 + fm;

    v8f acc = {0.f, 0.f, 0.f, 0.f, 0.f, 0.f, 0.f, 0.f};
    #pragma unroll
    for (int ks = 0; ks < KSTEPS; ++ks) {
        int k0 = ks * 4;
        v2f afrag = {s_coef[fm][k0 + fk], s_coef[fm][k0 + fk + 1]};
        v2f bfrag = {dcol[(k0 + fk) * NCOL], dcol[(k0 + fk + 1) * NCOL]};
        acc = __builtin_amdgcn_wmma_f32_16x16x4_f32(
            false, afrag, false, bfrag, (short)0, acc, false, false);
    }
    // C layout: VGPR r -> rows r (lanes 0-15) and r+8 (lanes 16-31), N = lane%16
    #pragma unroll
    for (int r = 0; r < 8; ++r) {
        int m = r + ((lane < 16) ? 0 : 8);
        s_vp[m][wave * 16 + fm] = acc[r];
    }

    // --- Stage 2: blend transforms with WMMA, waves split the 16 batches ---
    for (int bm = wave; bm < 16; bm += 3) {
        v8f tacc = {0.f, 0.f, 0.f, 0.f, 0.f, 0.f, 0.f, 0.f};
        const float* r2b = &s_R2[bm][0];
        #pragma unroll
        for (int ks = 0; ks < 4; ++ks) {
            int k0 = ks * 4;
            v2f afrag = {s_w[fm][k0 + fk], s_w[fm][k0 + fk + 1]};
            v2f bfrag = {(fm < 12) ? r2b[(k0 + fk) * 12 + fm] : 0.f,
                         (fm < 12) ? r2b[(k0 + fk + 1) * 12 + fm] : 0.f};
            tacc = __builtin_amdgcn_wmma_f32_16x16x4_f32(
                false, afrag, false, bfrag, (short)0, tacc, false, false);
        }
        #pragma unroll
        for (int r = 0; r < 8; ++r) {
            int m = r + ((lane < 16) ? 0 : 8);
            s_T[bm][m][fm] = tacc[r];
        }
    }
    __syncthreads();

    // --- Stage 3: apply blended transform, 256 (batch, vertex) pairs ---
    for (int idx = tid; idx < 256; idx += 96) {
        int bm = idx >> 4, vm = idx & 15;
        int v = vt * 16 + vm;
        size_t b = (size_t)bt * 16 + bm;
        if (v >= V_ || (int)b >= B) continue;
        float px = s_vp[bm][vm * 3 + 0];
        float py = s_vp[bm][vm * 3 + 1];
        float pz = s_vp[bm][vm * 3 + 2];
        const float* T = &s_T[bm][vm][0];
        float ox = T[0] * px + T[1] * py + T[2]  * pz + T[3];
        float oy = T[4] * px + T[5] * py + T[6]  * pz + T[7];
        float oz = T[8] * px + T[9] * py + T[10] * pz + T[11];
        float* o = out_verts + (b * V_ + v) * 3;
        o[0] = ox * 1000.f;
        o[1] = oy * 1000.f;
        o[2] = oz * 1000.f;
    }
}

// ---------------------------------------------------------------------------
// Assemble final 21-joint output (16 chain joints + 5 fingertips, reordered).
// ---------------------------------------------------------------------------
__global__ void jtr_kernel(const float* __restrict__ jtr16,
                           const float* __restrict__ verts,
                           float* __restrict__ jtr_out,
                           int B) {
    int t = blockIdx.x * blockDim.x + threadIdx.x;
    if (t >= B * 21) return;
    int b = t / 21, o = t % 21;
    int src = c_ORDER[o];
    float x, y, z;
    if (src < 16) {
        const float* p = jtr16 + ((size_t)b * 16 + src) * 3;
        x = p[0] * 1000.f; y = p[1] * 1000.f; z = p[2] * 1000.f;
    } else {
        int v = c_TIPS[src - 16];
        const float* p = verts + ((size_t)b * V_ + v) * 3;   // already x1000
        x = p[0]; y = p[1]; z = p[2];
    }
    float* po = jtr_out + (size_t)t * 3;
    po[0] = x; po[1] = y; po[2] = z;
}

// ---------------------------------------------------------------------------
extern "C" void kernel_launch(void* const* d_in, const int* in_sizes, int n_in,
                              void* d_out, int out_size, void* d_ws, size_t ws_size,
                              hipStream_t stream) {
    const float* pc    = (const float*)d_in[0];  // (B, 48)
    const float* betas = (const float*)d_in[1];  // (B, 10)
    const float* sdirs = (const float*)d_in[2];  // (V, 3, 10)
    const float* pdirs = (const float*)d_in[3];  // (V, 3, 135)
    const float* vtmpl = (const float*)d_in[4];  // (1, V, 3)
    const float* jreg  = (const float*)d_in[5];  // (16, V)
    const float* wgt   = (const float*)d_in[6];  // (V, 16)
    const float* comps = (const float*)d_in[7];  // (45, 45)
    const float* hmean = (const float*)d_in[8];  // (1, 45)
    float* out = (float*)d_out;

    int B = in_sizes[0] / 48;

    // workspace layout (floats)
    float* ws    = (float*)d_ws;
    float* Dir   = ws;                           // 148 * 2352
    float* JtJsd = Dir + (size_t)K_ * NCOL;      // 528
    float* coef  = JtJsd + 528;                  // B * 148
    float* R2    = coef + (size_t)B * K_;        // B * 192
    float* jtr16 = R2 + (size_t)B * 192;         // B * 48

    prep_dirs<<<(K_ * NCOL + 255) / 256, 256, 0, stream>>>(pdirs, sdirs, vtmpl, Dir);
    prep_jreg<<<1, 544, 0, stream>>>(jreg, sdirs, vtmpl, JtJsd);
    pose_kernel<<<(B + 255) / 256, 256, 0, stream>>>(pc, betas, comps, hmean, JtJsd,
                                                     coef, R2, jtr16, B);
    dim3 grid(VPAD / 16, (B + 15) / 16);
    mano_main<<<grid, 96, 0, stream>>>(Dir, coef, R2, wgt, out, B);

    float* jtr_out = out + (size_t)B * V_ * 3;
    jtr_kernel<<<(B * 21 + 255) / 256, 256, 0, stream>>>(jtr16, out, jtr_out, B);
}